// YoloLoss_3745211483093
// MI455X (gfx1250) — compile-verified
//
#include <hip/hip_runtime.h>
#include <cstdint>

typedef __attribute__((ext_vector_type(2))) float v2f;
typedef __attribute__((ext_vector_type(8))) float v8f;

#define WAVES_PER_BLOCK 4
#define THREADS_PER_BLOCK (WAVES_PER_BLOCK * 32)
#define FLOATS_PER_CELL 30
#define CELLS_PER_CHUNK 32
#define CHUNK_FLOATS (CELLS_PER_CHUNK * FLOATS_PER_CELL)  // 960
#define CHUNK_BYTES  (CHUNK_FLOATS * 4)                   // 3840

__device__ __forceinline__ float iou4(float gx, float gy, float gw, float gh,
                                      float bx, float by, float bw, float bh) {
    float dw = fabsf(gx - bx);
    float dh = fabsf(gy - by);
    float sw = (gw + bw) * 0.5f;
    float sh = (gh + bh) * 0.5f;
    float iw = fmaxf(sw - dw, 0.0f);
    float ih = fmaxf(sh - dh, 0.0f);
    float inter = iw * ih;
    float uni = gw * gh + bw * bh - inter;
    return inter / uni;
}

__device__ __forceinline__ float cell_loss(const float* __restrict__ o,
                                           const float* __restrict__ t) {
    float obj = t[20];
    float prob = 0.0f;
#pragma unroll
    for (int i = 0; i < 20; ++i) {
        float d = t[i] - o[i];
        prob += d * d;
    }
    float g0 = t[22], g1 = t[23], g2 = t[24], g3 = t[25];
    float a0 = o[22], a1 = o[23], a2 = o[24], a3 = o[25];
    float b0 = o[26], b1 = o[27], b2 = o[28], b3 = o[29];
    float iou1 = iou4(g0, g1, g2, g3, a0, a1, a2, a3);
    float iou2 = iou4(g0, g1, g2, g3, b0, b1, b2, b3);
    bool c1 = iou1 > iou2;
    float ch0 = c1 ? a0 : b0, ch1 = c1 ? a1 : b1;
    float ch2 = c1 ? a2 : b2, ch3 = c1 ? a3 : b3;
    float cr = c1 ? o[20] : o[21];
    float co = c1 ? o[21] : o[20];
    float dp0 = g0 - ch0, dp1 = g1 - ch1;
    float pos = dp0 * dp0 + dp1 * dp1;
    float ds2 = sqrtf(g2) - sqrtf(ch2);
    float ds3 = sqrtf(g3) - sqrtf(ch3);
    float sz = ds2 * ds2 + ds3 * ds3;
    float conf1 = (cr - 1.0f) * (cr - 1.0f);
    float conf2 = (co - 1.0f) * (co - 1.0f);
    float per = 5.0f * (pos + sz) + conf1 + 0.5f * conf2 + prob;
    return (obj != 0.0f) ? per : 0.0f;
}

// Issue the 16 async global->LDS copies for one 32-cell chunk (both matrices):
// 3840 B per matrix = 7 x (32 lanes x b128) + 1 x (32 lanes x b64), coalesced,
// tracked on ASYNCcnt.
__device__ __forceinline__ void issue_chunk_async(uint32_t cb, uint32_t lane,
                                                  uint32_t lds_o, uint32_t lds_t,
                                                  uint64_t gb_o, uint64_t gb_t) {
#pragma unroll
    for (uint32_t j = 0; j < 7; ++j) {
        uint32_t sub = j * 512u + lane * 16u;
        uint32_t vo = cb + sub;
        asm volatile("global_load_async_to_lds_b128 %0, %1, %2"
                     :: "v"(lds_o + sub), "v"(vo), "s"(gb_o) : "memory");
        asm volatile("global_load_async_to_lds_b128 %0, %1, %2"
                     :: "v"(lds_t + sub), "v"(vo), "s"(gb_t) : "memory");
    }
    {
        uint32_t sub = 3584u + lane * 8u;
        uint32_t vo = cb + sub;
        asm volatile("global_load_async_to_lds_b64 %0, %1, %2"
                     :: "v"(lds_o + sub), "v"(vo), "s"(gb_o) : "memory");
        asm volatile("global_load_async_to_lds_b64 %0, %1, %2"
                     :: "v"(lds_t + sub), "v"(vo), "s"(gb_t) : "memory");
    }
}

__global__ __launch_bounds__(THREADS_PER_BLOCK) void yolo_loss_main(
    const float* __restrict__ outputs, const float* __restrict__ targets,
    float* __restrict__ result, int n_chunks) {
    // [wave][buffer][matrix][floats] : 4 * 2 * 2 * 960 * 4B = 60 KB
    __shared__ float lds[WAVES_PER_BLOCK][2][2][CHUNK_FLOATS];

    const int lane = threadIdx.x & 31;
    const int wave = threadIdx.x >> 5;
    const int waveGlobal = blockIdx.x * WAVES_PER_BLOCK + wave;
    const int waveStride = gridDim.x * WAVES_PER_BLOCK;

    // Low 32 bits of a generic pointer to LDS == wave-relative LDS byte offset.
    uint32_t lds_buf_o[2], lds_buf_t[2];
    lds_buf_o[0] = (uint32_t)(uintptr_t)&lds[wave][0][0][0];
    lds_buf_t[0] = (uint32_t)(uintptr_t)&lds[wave][0][1][0];
    lds_buf_o[1] = (uint32_t)(uintptr_t)&lds[wave][1][0][0];
    lds_buf_t[1] = (uint32_t)(uintptr_t)&lds[wave][1][1][0];
    const uint64_t gb_o = (uint64_t)outputs;
    const uint64_t gb_t = (uint64_t)targets;

    float acc = 0.0f;
    int cur = 0;

    // Prologue: prefetch first chunk into buffer 0.
    if (waveGlobal < n_chunks) {
        issue_chunk_async((uint32_t)waveGlobal * CHUNK_BYTES, (uint32_t)lane,
                          lds_buf_o[0], lds_buf_t[0], gb_o, gb_t);
    }

    for (int chunk = waveGlobal; chunk < n_chunks; chunk += waveStride) {
        const int nxt = chunk + waveStride;
        const int oth = cur ^ 1;

        // WAR: reads of buffer `oth` from the previous iteration must drain
        // before the async engine overwrites it with the prefetch.
        asm volatile("s_wait_dscnt 0" ::: "memory");

        if (nxt < n_chunks) {
            // Prefetch next chunk into the other buffer, then wait only for
            // the *older* 16 copies (ASYNCcnt completes in order): overlap
            // next chunk's HBM traffic with this chunk's compute.
            issue_chunk_async((uint32_t)nxt * CHUNK_BYTES, (uint32_t)lane,
                              lds_buf_o[oth], lds_buf_t[oth], gb_o, gb_t);
            asm volatile("s_wait_asynccnt 16" ::: "memory");
        } else {
            asm volatile("s_wait_asynccnt 0" ::: "memory");
        }

        const float* o = &lds[wave][cur][0][lane * FLOATS_PER_CELL];
        const float* t = &lds[wave][cur][1][lane * FLOATS_PER_CELL];
        acc += cell_loss(o, t);

        cur = oth;
    }

    // Exact 32-lane f32 sum via V_WMMA_F32_16X16X4_F32 with B = ones:
    // A(m,0)=x_m (lanes 0-15 vgpr0), A(m,2)=x_{m+16} (lanes 16-31 vgpr0),
    // A(m,1)=A(m,3)=0  =>  D(m,n) = x_m + x_{m+16}  (replicated over n).
    v2f a;
    a[0] = acc;
    a[1] = 0.0f;
    v2f b;
    b[0] = 1.0f;
    b[1] = 1.0f;
    v8f c = {};
    v8f d = __builtin_amdgcn_wmma_f32_16x16x4_f32(
        /*neg_a=*/false, a, /*neg_b=*/false, b,
        /*c_mod=*/(short)0, c, /*reuse_a=*/false, /*reuse_b=*/false);

    // Lanes 0-15 hold rows m=0..7 in d[0..7]; lanes 16-31 hold rows m=8..15.
    float half = d[0] + d[1] + d[2] + d[3] + d[4] + d[5] + d[6] + d[7];
    float tot = half + __shfl_xor(half, 16, 32);

    if (lane == 0) atomicAdd(result, tot);
}

__global__ void yolo_loss_tail(const float* __restrict__ outputs,
                               const float* __restrict__ targets,
                               float* __restrict__ result, int start, int cells) {
    int i = start + (int)threadIdx.x;
    if (i < cells) {
        float v = cell_loss(outputs + (size_t)i * FLOATS_PER_CELL,
                            targets + (size_t)i * FLOATS_PER_CELL);
        atomicAdd(result, v);
    }
}

extern "C" void kernel_launch(void* const* d_in, const int* in_sizes, int n_in,
                              void* d_out, int out_size, void* d_ws, size_t ws_size,
                              hipStream_t stream) {
    const float* outputs = (const float*)d_in[0];
    const float* targets = (const float*)d_in[1];
    float* result = (float*)d_out;

    const long total = (long)in_sizes[0];              // 16384*7*7*30
    const int cells = (int)(total / FLOATS_PER_CELL);  // 802816
    const int chunks = cells / CELLS_PER_CHUNK;        // 25088
    const int tail_start = chunks * CELLS_PER_CHUNK;

    hipMemsetAsync(result, 0, sizeof(float), stream);

    // 1568 blocks x 4 waves = 6272 waves -> exactly 4 chunks per wave,
    // each double-buffered through its private 15 KB LDS slice.
    yolo_loss_main<<<1568, THREADS_PER_BLOCK, 0, stream>>>(outputs, targets,
                                                           result, chunks);
    if (tail_start < cells) {
        yolo_loss_tail<<<1, 32, 0, stream>>>(outputs, targets, result,
                                             tail_start, cells);
    }
}